// IPLS_48120813584730
// MI455X (gfx1250) — compile-verified
//
#include <hip/hip_runtime.h>
#include <math.h>

// ---------------- problem constants ----------------
#define NF 131072      // n_feat
#define NT 2048        // n_target
#define NL 64          // n_latent
#define GWG 64         // workgroups (grid)
#define BTH 256        // threads per workgroup (8 wave32 waves)
#define TOT (GWG*BTH)  // 16384 threads total
#define CHK (NF/TOT)   // 8 xc elements per thread
#define YCHK (NT/BTH)  // 8 yc elements per thread (full yc replicated per WG)
#define EPSF 1e-7f

typedef __attribute__((ext_vector_type(2))) float v2f;
typedef __attribute__((ext_vector_type(8))) float v8f;

// ---------------- init: zero the grid-barrier state each launch ----------------
__global__ void ipls_init_kernel(unsigned* sync) {
    if (threadIdx.x < 2) sync[threadIdx.x] = 0u;
}

// ---------------- software grid barrier (sense-reversal via generation) ----------------
__device__ __forceinline__ void grid_barrier(unsigned* cnt, unsigned* gen) {
    __syncthreads();
    if (threadIdx.x == 0) {
        __threadfence();  // make this WG's partial-sum stores device-visible
        unsigned g = __hip_atomic_load(gen, __ATOMIC_RELAXED, __HIP_MEMORY_SCOPE_AGENT);
        unsigned prev = __hip_atomic_fetch_add(cnt, 1u, __ATOMIC_ACQ_REL, __HIP_MEMORY_SCOPE_AGENT);
        if (prev == (unsigned)(GWG - 1)) {
            __hip_atomic_store(cnt, 0u, __ATOMIC_RELAXED, __HIP_MEMORY_SCOPE_AGENT);
            __hip_atomic_fetch_add(gen, 1u, __ATOMIC_RELEASE, __HIP_MEMORY_SCOPE_AGENT);
        } else {
            while (__hip_atomic_load(gen, __ATOMIC_ACQUIRE, __HIP_MEMORY_SCOPE_AGENT) == g)
                __builtin_amdgcn_s_sleep(2);
        }
        __threadfence();
    }
    __syncthreads();
}

// ---------------- WMMA-based workgroup reduction of 6 arrays of 256 floats ----------------
// Six waves each reduce one array: 4x V_WMMA_F32_16X16X4_F32 against an all-ones B
// fold 256 values into 16 row-sums (fp32-exact layout per ISA 7.12.2), then 8 per-lane
// adds + one shfl_xor(16) produce the total in every lane.
__device__ __forceinline__ void wg_reduce6(float p0, float p1, float p2,
                                           float p3, float p4, float p5,
                                           float red[6][BTH], float rres[6]) {
    const int tid = threadIdx.x;
    red[0][tid] = p0; red[1][tid] = p1; red[2][tid] = p2;
    red[3][tid] = p3; red[4][tid] = p4; red[5][tid] = p5;
    __syncthreads();
    const int wv   = tid >> 5;   // wave id (uniform per wave -> EXEC stays all-ones)
    const int lane = tid & 31;
    if (wv < 6) {
        const float* s = red[wv];
        const int m  = lane & 15;            // A-matrix row
        const int kh = (lane >> 4) << 1;     // K half: lanes 0-15 -> K0/K1, lanes 16-31 -> K2/K3
        v2f bones; bones[0] = 1.0f; bones[1] = 1.0f;          // all-ones B (layout-independent)
        v8f acc = {0.f, 0.f, 0.f, 0.f, 0.f, 0.f, 0.f, 0.f};
        #pragma unroll
        for (int c = 0; c < 4; ++c) {                          // 4 chunks of 64 values
            v2f a;
            a[0] = s[c * 64 + m * 4 + kh + 0];                 // A[m, kh]
            a[1] = s[c * 64 + m * 4 + kh + 1];                 // A[m, kh+1]
            acc = __builtin_amdgcn_wmma_f32_16x16x4_f32(
                false, a, false, bones, (short)0, acc, false, false);
        }
        // acc[j] = rowsum(M = j (+8 for upper lanes)), identical across N (all-ones B)
        float r = acc[0] + acc[1] + acc[2] + acc[3] + acc[4] + acc[5] + acc[6] + acc[7];
        r += __shfl_xor(r, 16, 32);                            // combine rows 0-7 with 8-15
        if (lane == 0) rres[wv] = r;
    }
    __syncthreads();
}

// ---------------- main persistent cooperative kernel ----------------
extern "C" __global__ __launch_bounds__(BTH)
void ipls_main_kernel(const float* __restrict__ x,   const float* __restrict__ y,
                      const float* __restrict__ mux, const float* __restrict__ muy,
                      const float* __restrict__ U,   const float* __restrict__ Wz,
                      const float* __restrict__ Cz,  const float* __restrict__ TSS,
                      const float* __restrict__ BZ,  const float* __restrict__ Pm,
                      const int*   __restrict__ nptr,
                      float* o_mux, float* o_muy, float* o_u, float* o_wz,
                      float* o_cz,  float* o_tss, float* o_bz, float* o_p,
                      unsigned* sync, float* gpart) {
    __shared__ float red[6][BTH];
    __shared__ float rres[6];
    __shared__ float bx[3 * GWG];
    __shared__ float sc[3];

    const int tid = threadIdx.x;
    const int wg  = blockIdx.x;
    const int gt  = wg * BTH + tid;

    // ---- prologue: running means + centered sample, all register-resident ----
    const float nf = (float)(*nptr);
    const float c0 = nf / (nf + 1.0f);
    const float c1 = 1.0f / (nf + 1.0f);

    float xc[CHK], yc[YCHK];
    #pragma unroll
    for (int k = 0; k < CHK; ++k) {
        const int idx = gt + k * TOT;                 // block-cyclic, fully coalesced
        const float m = mux[idx] * c0 + x[idx] * c1;
        o_mux[idx] = m;
        xc[k] = x[idx] - m;
    }
    #pragma unroll
    for (int k = 0; k < YCHK; ++k) {
        const int j = tid + k * BTH;
        const float m = muy[j] * c0 + y[j] * c1;
        if (wg == 0) o_muy[j] = m;
        yc[k] = y[j] - m;                             // full yc replicated per WG
    }

    // ---- sequential latent loop: 1 grid barrier per latent ----
    for (int i = 0; i < NL; ++i) {
        float wzr[CHK], czr[YCHK];
        float pA = 0.f, pB = 0.f, pC = 0.f, pD = 0.f, pE = 0.f, pF = 0.f;

        #pragma unroll
        for (int k = 0; k < CHK; ++k) {
            const int idx = i * NF + gt + k * TOT;
            const float wv = Wz[idx];
            wzr[k] = wv;
            pA += xc[k] * wv;           // A = sum(xc*wzr)
            pB += xc[k] * xc[k];        // B = sum(xc^2)
            pC += wv * wv;              // C = sum(wzr^2)
        }
        if (i + 1 < NL) {               // hide next latent's row fetch behind this one
            __builtin_prefetch(&Wz[(i + 1) * NF + gt], 0, 1);
            __builtin_prefetch(&Pm[(i + 1) * NF + gt], 0, 1);
        }
        #pragma unroll
        for (int k = 0; k < YCHK; ++k) {
            const float cc = Cz[i * NT + tid + k * BTH];
            czr[k] = cc;
            pD += yc[k] * cc;           // D = sum(yc*czr)
            pE += yc[k] * yc[k];        // E = sum(yc^2)
            pF += cc * cc;              // F = sum(czr^2)
        }

        wg_reduce6(pA, pB, pC, pD, pE, pF, red, rres);

        // publish x-side partials (double-buffered by latent parity)
        if (tid == 0) {
            float* slot = gpart + (((i & 1) * GWG) + wg) * 3;
            slot[0] = rres[0]; slot[1] = rres[1]; slot[2] = rres[2];
        }
        const float Dv = rres[3], Ev = rres[4], Fv = rres[5];  // y-side: WG-local, exact

        grid_barrier(sync, sync + 1);

        // deterministic cross-WG combine (fixed serial order)
        if (tid < GWG) {
            const volatile float* slot = gpart + (((i & 1) * GWG) + tid) * 3;
            bx[tid]           = slot[0];
            bx[GWG + tid]     = slot[1];
            bx[2 * GWG + tid] = slot[2];
        }
        __syncthreads();
        if (tid == 0) {
            float a = 0.f, b = 0.f, c = 0.f;
            for (int t = 0; t < GWG; ++t) {
                a += bx[t]; b += bx[GWG + t]; c += bx[2 * GWG + t];
            }
            sc[0] = a; sc[1] = b; sc[2] = c;
        }
        __syncthreads();
        const float Av = sc[0], Bv = sc[1], Cv = sc[2];

        // ---- scalar chain: both inner passes via closed forms in A..F ----
        const float u0 = U[i], tss0 = TSS[i];
        // pass 1 (wz = wzr + u0*xc)
        const float Sxw1 = Av + u0 * Bv;
        const float Sww1 = Cv + 2.0f * u0 * Av + u0 * u0 * Bv;
        const float tz1  = Sxw1 / (sqrtf(Sww1) + EPSF);
        const float tss1 = tss0 + tz1 * tz1;
        const float t1   = tz1 / sqrtf(tss1);
        const float G1   = Fv + 2.0f * t1 * Dv + t1 * t1 * Ev;   // sum(cz1^2)
        const float u1   = (Dv + t1 * Ev) / sqrtf(G1);           // dot(yc, C_i)
        // pass 2 (wz = wzr + u1*xc)  -- uses original tss0 per reference
        const float Sxw2 = Av + u1 * Bv;
        const float Sww2 = Cv + 2.0f * u1 * Av + u1 * u1 * Bv;
        const float tz2  = Sxw2 / (sqrtf(Sww2) + EPSF);
        const float tss2 = tss0 + tz2 * tz2;
        const float t2   = tz2 / sqrtf(tss2);
        const float G2   = Fv + 2.0f * t2 * Dv + t2 * t2 * Ev;   // sum(cz2^2)
        const float u2   = (Dv + t2 * Ev) / sqrtf(G2);
        const float bzn  = BZ[i] + u2 * tz2;
        const float bb   = bzn / sqrtf(tss2);
        const float kk   = bb * t2 / sqrtf(G2);                  // yc -= kk * cz2

        // ---- vector tail: write Wz/P rows, deflate xc ----
        #pragma unroll
        for (int k = 0; k < CHK; ++k) {
            const int idx = i * NF + gt + k * TOT;
            o_wz[idx] = wzr[k] + xc[k] * u1;          // wz of pass 2 uses u after pass 1
            const float pn = Pm[idx] + xc[k] * t2;
            o_p[idx] = pn;
            xc[k] -= t2 * pn;                          // deflate x
        }
        #pragma unroll
        for (int k = 0; k < YCHK; ++k) {
            const float cz2 = czr[k] + yc[k] * t2;
            if (wg == 0) o_cz[i * NT + tid + k * BTH] = cz2;
            yc[k] -= kk * cz2;                         // deflate y (identical in every WG)
        }
        if (wg == 0 && tid == 0) {
            o_u[i] = u2; o_tss[i] = tss2; o_bz[i] = bzn;
        }
    }
}

// ---------------- host launcher ----------------
extern "C" void kernel_launch(void* const* d_in, const int* in_sizes, int n_in,
                              void* d_out, int out_size, void* d_ws, size_t ws_size,
                              hipStream_t stream) {
    (void)in_sizes; (void)n_in; (void)out_size; (void)ws_size;
    const float* x    = (const float*)d_in[0];
    const float* y    = (const float*)d_in[1];
    const float* mux  = (const float*)d_in[2];
    const float* muy  = (const float*)d_in[3];
    const float* U    = (const float*)d_in[4];
    const float* Wz   = (const float*)d_in[5];
    const float* Cz   = (const float*)d_in[6];
    const float* TSS  = (const float*)d_in[7];
    const float* BZ   = (const float*)d_in[8];
    const float* Pm   = (const float*)d_in[9];
    const int*   nptr = (const int*)d_in[10];

    float* o = (float*)d_out;
    float* o_mux = o;                         // 131072
    float* o_muy = o + NF;                    // 2048
    float* o_u   = o + NF + NT;               // 64
    float* o_wz  = o_u + NL;                  // 64*131072
    float* o_cz  = o_wz + (size_t)NL * NF;    // 64*2048
    float* o_tss = o_cz + (size_t)NL * NT;    // 64
    float* o_bz  = o_tss + NL;                // 64
    float* o_p   = o_bz + NL;                 // 64*131072

    unsigned* sync  = (unsigned*)d_ws;                       // [cnt, gen]
    float*    gpart = (float*)((char*)d_ws + 256);           // 2 x GWG x 3 floats

    ipls_init_kernel<<<1, 32, 0, stream>>>(sync);
    ipls_main_kernel<<<GWG, BTH, 0, stream>>>(x, y, mux, muy, U, Wz, Cz, TSS, BZ, Pm,
                                              nptr, o_mux, o_muy, o_u, o_wz, o_cz,
                                              o_tss, o_bz, o_p, sync, gpart);
}